// DifferentiableDenoising_67860483277351
// MI455X (gfx1250) — compile-verified
//
#include <hip/hip_runtime.h>
#include <stdint.h>

// Bilateral denoise: [2,512,512,3] fp32, 9x9 window, sigma_color=0.1, sigma_space=3,
// out = 0.8*x + 0.2*bilateral(x), clamped to [0,1].
//
// MI455X notes: problem is ~12.6MB of traffic (0.5us at 23.3TB/s) but ~42M exp taps
// -> TRANS/VALU bound. No matmul structure (per-pixel data-dependent weights), so no
// WMMA; we use the CDNA5 async global->LDS copy path + asynccnt waits + LDS tiling.

#define IMG_H 512
#define IMG_W 512
#define NCH 3
#define TILE 32
#define HPAD 4
#define LW (TILE + 2 * HPAD)   // 40 padded rows/cols
#define LSTRIDE 40             // floats per LDS row (lanes stride 1 -> conflict free)

__device__ __forceinline__ int reflect512(int p) {
    // jnp.pad 'reflect' (no edge repeat); single fold suffices for p in [-4, 515]
    p = (p < 0) ? -p : p;
    p = (p >= 512) ? (1022 - p) : p;
    return p;
}

__launch_bounds__(256)
__global__ void bilateral_denoise_kernel(const float* __restrict__ img,
                                         float* __restrict__ out)
{
    __shared__ float lds[NCH][LW * LSTRIDE];   // 3 * 40*40 * 4B = 18.75 KB

    const int tid   = threadIdx.x;
    const int tileX = blockIdx.x;
    const int tileY = blockIdx.y;
    const int b     = blockIdx.z;
    const int x0    = tileX * TILE;
    const int y0    = tileY * TILE;

    // ------------------------------------------------------------------
    // Stage padded 40x40 tile (3 channel planes, SoA) into LDS using the
    // CDNA5 async global->LDS copy path. Per-lane reflect-mapped byte
    // offset (VADDR) + uniform base (SADDR) -> GVS addressing. Tracked by
    // ASYNCcnt; completion enforced with s_wait_asynccnt before barrier.
    // ------------------------------------------------------------------
    const uint64_t base = (uint64_t)(uintptr_t)img;
    for (int e = tid; e < LW * LW; e += 256) {
        const int ly = e / LW;
        const int lx = e - ly * LW;
        const int gy = reflect512(y0 + ly - HPAD);
        const int gx = reflect512(x0 + lx - HPAD);
        const uint32_t goff =
            (uint32_t)((((b * IMG_H + gy) * IMG_W + gx) * NCH) * (int)sizeof(float));
        const uint32_t l0 = (uint32_t)(uintptr_t)&lds[0][ly * LSTRIDE + lx];
        const uint32_t l1 = (uint32_t)(uintptr_t)&lds[1][ly * LSTRIDE + lx];
        const uint32_t l2 = (uint32_t)(uintptr_t)&lds[2][ly * LSTRIDE + lx];
        // NOTE: inst offset applies to BOTH lds and mem addresses, so channel
        // strides (4B global vs 6400B lds) are folded into the operands instead.
        asm volatile("global_load_async_to_lds_b32 %0, %1, %2"
                     :: "v"(l0), "v"(goff),      "s"(base) : "memory");
        asm volatile("global_load_async_to_lds_b32 %0, %1, %2"
                     :: "v"(l1), "v"(goff + 4u), "s"(base) : "memory");
        asm volatile("global_load_async_to_lds_b32 %0, %1, %2"
                     :: "v"(l2), "v"(goff + 8u), "s"(base) : "memory");
    }
    asm volatile("s_wait_asynccnt 0" ::: "memory");
    __syncthreads();

    // ------------------------------------------------------------------
    // Compute: each thread owns 4 vertically consecutive output pixels at
    // column tx; slide over the 12 shared input rows so each LDS row
    // segment (9 cols x 3 ch) is loaded once and reused by up to 4 pixels.
    // ------------------------------------------------------------------
    const int tx  = tid & 31;
    const int ty0 = (tid >> 5) * 4;

    // normalized 1D gaussian, sigma=3, K=9 (matches kornia/reference)
    const float Wsp[9] = {0.06297013f, 0.09290260f, 0.12264923f, 0.14489292f,
                          0.15317031f,
                          0.14489292f, 0.12264923f, 0.09290260f, 0.06297013f};

    float cr[4], cg[4], cb[4];
    float nr[4], ng[4], nb[4], den[4];
#pragma unroll
    for (int k = 0; k < 4; ++k) {
        const int ci = (ty0 + k + HPAD) * LSTRIDE + (tx + HPAD);
        cr[k] = lds[0][ci];
        cg[k] = lds[1][ci];
        cb[k] = lds[2][ci];
        nr[k] = ng[k] = nb[k] = den[k] = 0.0f;
    }

#pragma unroll
    for (int iy = 0; iy < 12; ++iy) {
        const int rb = (ty0 + iy) * LSTRIDE + tx;
        float pr[9], pg[9], pb[9];
#pragma unroll
        for (int dx = 0; dx < 9; ++dx) {
            pr[dx] = lds[0][rb + dx];
            pg[dx] = lds[1][rb + dx];
            pb[dx] = lds[2][rb + dx];
        }
#pragma unroll
        for (int k = 0; k < 4; ++k) {
            const int dy = iy - k;               // compile-time after unroll
            if (dy < 0 || dy > 8) continue;      // folds away
            const float wy = Wsp[dy];
#pragma unroll
            for (int dx = 0; dx < 9; ++dx) {
                const float dr = pr[dx] - cr[k];
                const float dg = pg[dx] - cg[k];
                const float db = pb[dx] - cb[k];
                const float d2 = dr * dr + dg * dg + db * db;
                // -0.5 / sigma_color^2 = -50 ; lowers to one v_exp_f32
                const float w = wy * Wsp[dx] * __expf(-50.0f * d2);
                nr[k]  += w * pr[dx];
                ng[k]  += w * pg[dx];
                nb[k]  += w * pb[dx];
                den[k] += w;
            }
        }
    }

#pragma unroll
    for (int k = 0; k < 4; ++k) {
        const int gy = y0 + ty0 + k;
        const int gx = x0 + tx;
        const size_t o = (((size_t)b * IMG_H + gy) * IMG_W + gx) * NCH;
        const float inv = 1.0f / den[k];   // den >= center weight > 0
        float r = 0.8f * cr[k] + 0.2f * (nr[k] * inv);
        float g = 0.8f * cg[k] + 0.2f * (ng[k] * inv);
        float bl = 0.8f * cb[k] + 0.2f * (nb[k] * inv);
        out[o + 0] = fminf(fmaxf(r, 0.0f), 1.0f);
        out[o + 1] = fminf(fmaxf(g, 0.0f), 1.0f);
        out[o + 2] = fminf(fmaxf(bl, 0.0f), 1.0f);
    }
}

extern "C" void kernel_launch(void* const* d_in, const int* in_sizes, int n_in,
                              void* d_out, int out_size, void* d_ws, size_t ws_size,
                              hipStream_t stream) {
    (void)in_sizes; (void)n_in; (void)d_ws; (void)ws_size; (void)out_size;
    const float* img = (const float*)d_in[0];
    float* out = (float*)d_out;
    dim3 grid(IMG_W / TILE, IMG_H / TILE, 2);   // 16 x 16 x 2 = 512 workgroups
    dim3 block(256);                            // 8 wave32 waves per WGP slot
    bilateral_denoise_kernel<<<grid, block, 0, stream>>>(img, out);
}